// Autoencoder_137438953935
// MI455X (gfx1250) — compile-verified
//
#include <hip/hip_runtime.h>
#include <math.h>

#define BATCH 16
#define SEQ   512
#define DIN   4096
#define HID   4096
#define NCOMP 8192
#define TOPK  256
#define CHUNK 64                          // K-rows per TDM transfer
#define SQRT_NCOMP 90.50966799187809f     // sqrt(8192)

typedef float v2f  __attribute__((ext_vector_type(2)));
typedef float v8f  __attribute__((ext_vector_type(8)));
typedef unsigned int u32x4 __attribute__((ext_vector_type(4)));
typedef int   i32x8 __attribute__((ext_vector_type(8)));
typedef int   i32x4 __attribute__((ext_vector_type(4)));

// ---------------------------------------------------------------------------
// Stage 1: pooled[b,d] = mean_s x[b,s,d].   134 MB stream, float4-vectorized.
// ---------------------------------------------------------------------------
__global__ __launch_bounds__(256) void pool_kernel(const float* __restrict__ x,
                                                   float* __restrict__ pooled,
                                                   float* __restrict__ out_pooled)
{
    const int d4 = blockIdx.x * blockDim.x + threadIdx.x;
    const int b  = blockIdx.y;
    const float4* xp = reinterpret_cast<const float4*>(x) +
                       (size_t)b * (SEQ * (DIN / 4)) + d4;
    float sx = 0.f, sy = 0.f, sz = 0.f, sw = 0.f;
#pragma unroll 4
    for (int s = 0; s < SEQ; ++s) {
        float4 v = xp[(size_t)s * (DIN / 4)];
        sx += v.x; sy += v.y; sz += v.z; sw += v.w;
    }
    const float inv = 1.0f / (float)SEQ;
    float4 r; r.x = sx * inv; r.y = sy * inv; r.z = sz * inv; r.w = sw * inv;
    reinterpret_cast<float4*>(pooled)    [(size_t)b * (DIN / 4) + d4] = r;
    reinterpret_cast<float4*>(out_pooled)[(size_t)b * (DIN / 4) + d4] = r;
}

// ---------------------------------------------------------------------------
// Skinny GEMM with TDM staging:
//   C[16,N] = act(A[16,K] @ W[K,N] + biasScale*bias)
// One wave32 per 16-column tile. W tile rows are pulled [CHUNK x 16] at a
// time by the Tensor Data Mover into double-buffered LDS (D# built in
// SGPR-uniform fields), synchronized on TENSORcnt; math is
// V_WMMA_F32_16X16X4_F32 fed by conflict-free ds_load_b32 for B and an
// aligned b64 global load for the (L2-resident) A fragment.
// ---------------------------------------------------------------------------
template <int RELU>
__global__ __launch_bounds__(256) void gemm16_wmma(const float* __restrict__ A,
                                                   const float* __restrict__ W,
                                                   const float* __restrict__ bias,
                                                   float* __restrict__ C,
                                                   int K, int N, float biasScale)
{
    __shared__ float lds[8][2][CHUNK * 16];            // 8 waves * 2 * 4KB = 64KB

    const int lane = threadIdx.x & 31;
    const int wave = threadIdx.x >> 5;
    const int tile = blockIdx.x * 8 + wave;
    const int n0   = tile << 4;
    if (n0 >= N) return;                               // wave-uniform guard

    const int half = lane >> 4;
    const int l    = lane & 15;

    const float* Ap = A + (size_t)l * K + (half << 1);
    __builtin_prefetch(Ap, 0, 1);                      // global_prefetch_b8

    // LDS byte offset of this wave's staging buffers (block LDS base == 0)
    const unsigned ldsBase =
        (unsigned)((const char*)&lds[wave][0][0] - (const char*)&lds[0][0][0]);

    // ---- TDM descriptor issue: one [CHUNK x 16] f32 tile of W ----
    auto tdm_issue = [&](int c, int buf) {
        const unsigned long long ga =
            (unsigned long long)(uintptr_t)(W + (size_t)c * CHUNK * N + n0);
        u32x4 g0;
        g0[0] = 1u;                                        // count=1, user mode
        g0[1] = ldsBase + (unsigned)buf * (CHUNK * 16 * 4);// lds_addr (bytes)
        g0[2] = (unsigned)ga;                              // global_addr[31:0]
        g0[3] = ((unsigned)(ga >> 32) & 0x01FFFFFFu)       // global_addr[56:32]
                | 0x80000000u;                             // type=2 ("image")
        i32x8 g1;
        g1[0] = (int)0x00020000;        // data_size=2 (4 bytes), no mask/pad
        g1[1] = (int)(16u << 16);       // tensor_dim0 = 16 cols
        g1[2] = (int)((unsigned)CHUNK << 16);  // tensor_dim1 = 64 rows
        g1[3] = (int)(16u << 16);       // tile_dim0 = 16
        g1[4] = (int)CHUNK;             // tile_dim1 = 64
        g1[5] = (int)N;                 // tensor_dim0_stride = row pitch
        g1[6] = 0;
        g1[7] = 0;
        i32x4 z4 = {0, 0, 0, 0};        // groups 2/3 unused (2-D tensor)
        i32x8 z8 = {0, 0, 0, 0, 0, 0, 0, 0};
        __builtin_amdgcn_tensor_load_to_lds(g0, g1, z4, z4, z8, 0);
    };

    v8f acc = {};
    const int nch = K / CHUNK;
    tdm_issue(0, 0);

    size_t k0 = 0;
    for (int c = 0; c < nch; ++c) {
        const int buf = c & 1;
        if (c + 1 < nch) {
            // buffer buf^1 was consumed one chunk ago; drain DS reads, refill
            asm volatile("s_wait_dscnt 0x0" ::: "memory");
            tdm_issue(c + 1, buf ^ 1);
            __builtin_amdgcn_s_wait_tensorcnt(1);  // oldest (current buf) done
        } else {
            __builtin_amdgcn_s_wait_tensorcnt(0);
        }
        asm volatile("" ::: "memory");             // no LDS reads above waits

        const float* Lb = &lds[wave][buf][0];
#pragma unroll
        for (int kk = 0; kk < CHUNK; kk += 4) {
            v2f a = *reinterpret_cast<const v2f*>(Ap + k0 + kk);
            v2f b;
            b.x = Lb[(kk + (half << 1)) * 16 + l];
            b.y = Lb[(kk + (half << 1) + 1) * 16 + l];
            acc = __builtin_amdgcn_wmma_f32_16x16x4_f32(
                /*neg_a=*/false, a, /*neg_b=*/false, b,
                /*c_mod=*/(short)0, acc, /*reuse_a=*/false, /*reuse_b=*/false);
        }
        k0 += CHUNK;
    }

    const float bv = bias[n0 + l] * biasScale;
    float* cp = C + (size_t)(half << 3) * N + n0 + l;
#pragma unroll
    for (int r = 0; r < 8; ++r) {
        float v = acc[r] + bv;
        if (RELU) v = fmaxf(v, 0.0f);
        cp[(size_t)r * N] = v;
    }
}

// ---------------------------------------------------------------------------
// Stage 4: per batch row -- softmax stats, exact top-256 (descending,
// lower-index tie-break like jax.lax.top_k), fused points = w * components.
// ---------------------------------------------------------------------------
__global__ __launch_bounds__(256) void softmax_topk_points(
    const float* __restrict__ logits,      // [BATCH, NCOMP]
    const float* __restrict__ components,  // [NCOMP, 3]
    float* __restrict__ points_out)        // [BATCH, TOPK, 3]
{
    __shared__ float vals[NCOMP];
    __shared__ float sred[256];
    __shared__ int   sidx[256];

    const int b   = blockIdx.x;
    const int tid = threadIdx.x;
    const float* row = logits + (size_t)b * NCOMP;

    float lmax = -INFINITY;
    for (int i = tid; i < NCOMP; i += 256) {
        float v = row[i];
        vals[i] = v;
        lmax = fmaxf(lmax, v);
    }
    sred[tid] = lmax;
    __syncthreads();
    for (int s = 128; s > 0; s >>= 1) {
        if (tid < s) sred[tid] = fmaxf(sred[tid], sred[tid + s]);
        __syncthreads();
    }
    const float rowmax = sred[0];
    __syncthreads();

    float lsum = 0.f;
    for (int i = tid; i < NCOMP; i += 256) lsum += expf(vals[i] - rowmax);
    sred[tid] = lsum;
    __syncthreads();
    for (int s = 128; s > 0; s >>= 1) {
        if (tid < s) sred[tid] += sred[tid + s];
        __syncthreads();
    }
    const float scale = SQRT_NCOMP / sred[0];
    __syncthreads();

    for (int k = 0; k < TOPK; ++k) {
        float bv = -INFINITY;
        int   bi = NCOMP;
        for (int i = tid; i < NCOMP; i += 256) {
            float v = vals[i];
            if (v > bv) { bv = v; bi = i; }    // strict '>' keeps lowest index
        }
        sred[tid] = bv; sidx[tid] = bi;
        __syncthreads();
        for (int s = 128; s > 0; s >>= 1) {
            if (tid < s) {
                float ov = sred[tid + s]; int oi = sidx[tid + s];
                if (ov > sred[tid] || (ov == sred[tid] && oi < sidx[tid])) {
                    sred[tid] = ov; sidx[tid] = oi;
                }
            }
            __syncthreads();
        }
        if (tid == 0) {
            const int   idx = sidx[0];
            const float w   = expf(sred[0] - rowmax) * scale;
            float* po = points_out + ((size_t)b * TOPK + k) * 3;
            po[0] = w * components[idx * 3 + 0];
            po[1] = w * components[idx * 3 + 1];
            po[2] = w * components[idx * 3 + 2];
            vals[idx] = -INFINITY;
        }
        __syncthreads();
    }
}

// ---------------------------------------------------------------------------
// Stage 5: g[b,j] = sum_k relu(points[b,k,:] . Wd1[:,j] + bd1[j])
// (sum over K commutes past the Wd2 matmul -> kills the 137-GFLOP GEMM)
// ---------------------------------------------------------------------------
__global__ __launch_bounds__(256) void decoder_hidden_sum(
    const float* __restrict__ points,  // [BATCH, TOPK, 3]
    const float* __restrict__ Wd1,     // [3, HID]
    const float* __restrict__ bd1,     // [HID]
    float* __restrict__ g)             // [BATCH, HID]
{
    __shared__ float p[TOPK * 3];
    const int b = blockIdx.y;
    const int j = blockIdx.x * 256 + threadIdx.x;
    for (int i = threadIdx.x; i < TOPK * 3; i += 256)
        p[i] = points[(size_t)b * TOPK * 3 + i];
    __syncthreads();

    const float w0 = Wd1[j];
    const float w1 = Wd1[HID + j];
    const float w2 = Wd1[2 * HID + j];
    const float bb = bd1[j];
    float acc = 0.f;
#pragma unroll 4
    for (int k = 0; k < TOPK; ++k)
        acc += fmaxf(p[3 * k] * w0 + p[3 * k + 1] * w1 + p[3 * k + 2] * w2 + bb, 0.f);
    g[(size_t)b * HID + j] = acc;
}

// ---------------------------------------------------------------------------
extern "C" void kernel_launch(void* const* d_in, const int* in_sizes, int n_in,
                              void* d_out, int out_size, void* d_ws, size_t ws_size,
                              hipStream_t stream)
{
    const float* x    = (const float*)d_in[0];
    const float* W1   = (const float*)d_in[1];
    const float* b1   = (const float*)d_in[2];
    const float* W2   = (const float*)d_in[3];
    const float* b2   = (const float*)d_in[4];
    const float* comp = (const float*)d_in[5];
    const float* Wd1  = (const float*)d_in[6];
    const float* bd1  = (const float*)d_in[7];
    const float* Wd2  = (const float*)d_in[8];
    const float* bd2  = (const float*)d_in[9];

    // output layout: points [16*256*3] | decoded [16*4096] | pooled [16*4096]
    float* out         = (float*)d_out;
    float* points_out  = out;
    float* decoded_out = out + BATCH * TOPK * 3;
    float* pooled_out  = out + BATCH * TOPK * 3 + BATCH * DIN;

    // workspace
    float* ws     = (float*)d_ws;
    float* pooled = ws;                          // 16*4096
    float* h      = pooled + BATCH * DIN;        // 16*4096
    float* logits = h + BATCH * HID;             // 16*8192
    float* g      = logits + BATCH * NCOMP;      // 16*4096

    // 1) mean over seq
    pool_kernel<<<dim3(DIN / 4 / 256, BATCH), 256, 0, stream>>>(x, pooled, pooled_out);

    // 2) h = relu(pooled @ W1 + b1)       : 256 tiles, 8 waves/block
    gemm16_wmma<1><<<dim3(HID / 16 / 8), 256, 0, stream>>>(
        pooled, W1, b1, h, DIN, HID, 1.0f);

    // 3) logits = h @ W2 + b2             : 512 tiles
    gemm16_wmma<0><<<dim3(NCOMP / 16 / 8), 256, 0, stream>>>(
        h, W2, b2, logits, HID, NCOMP, 1.0f);

    // 4) softmax scale + exact top-256 + points
    softmax_topk_points<<<dim3(BATCH), 256, 0, stream>>>(logits, comp, points_out);

    // 5) g = sum_k relu(points @ Wd1 + bd1)
    decoder_hidden_sum<<<dim3(HID / 256, BATCH), 256, 0, stream>>>(
        points_out, Wd1, bd1, g);

    // 6) decoded = g @ Wd2 + 256*bd2  (bias summed K times per reference)
    gemm16_wmma<0><<<dim3(DIN / 16 / 8), 256, 0, stream>>>(
        g, Wd2, bd2, decoded_out, HID, DIN, (float)TOPK);
}